// DistributeLayer_14800457302561
// MI455X (gfx1250) — compile-verified
//
#include <hip/hip_runtime.h>
#include <hip/hip_bf16.h>

typedef __attribute__((ext_vector_type(2))) float v2f;
typedef __attribute__((ext_vector_type(4))) float v4f;
typedef __attribute__((ext_vector_type(8))) float v8f;

#define BS    4
#define LENM  256
#define LENE  512
#define HDIM  512
#define VOCAB 32000

// ---------------------------------------------------------------------------
// Zero-fill output with non-temporal 128-bit stores (don't pollute L2 with 0s)
// ---------------------------------------------------------------------------
__global__ void zero_out_kernel(float* __restrict__ out, long n4) {
  long i = (long)blockIdx.x * blockDim.x + threadIdx.x;
  long stride = (long)gridDim.x * blockDim.x;
  v4f z = {0.f, 0.f, 0.f, 0.f};
  v4f* p = (v4f*)out;
  for (; i < n4; i += stride) {
    __builtin_nontemporal_store(z, p + i);
  }
}

// ---------------------------------------------------------------------------
// Y[r][n] = scale * sum_k X[r][k] * W[k][n]
// X: rows x 512 row-major, W: 512 x 512 row-major.
// One wave -> 16x64 output strip (4 WMMA accumulators), block = 4 waves.
// grid = (rows/16, 512/256)
// ---------------------------------------------------------------------------
__global__ void proj_gemm_wmma(const float* __restrict__ X,
                               const float* __restrict__ W,
                               float* __restrict__ Y,
                               float scale) {
  const int lane = threadIdx.x & 31;
  const int wave = threadIdx.x >> 5;
  const int row0 = blockIdx.x * 16;
  const int col0 = blockIdx.y * 256 + wave * 64;
  const int hi   = lane >> 4;   // 0: K pair {0,1}, 1: K pair {2,3}
  const int lo   = lane & 15;   // A: row-in-tile, B/D: col-in-tile

  v8f acc0 = {}, acc1 = {}, acc2 = {}, acc3 = {};

  const float* xrow = X + (size_t)(row0 + lo) * HDIM;
  for (int k0 = 0; k0 < HDIM; k0 += 4) {
    const int kk = k0 + 2 * hi;
    v2f a;
    a.x = xrow[kk];
    a.y = xrow[kk + 1];

    v2f b0, b1, b2, b3;
    const float* w0 = W + (size_t)kk * HDIM + col0 + lo;
    const float* w1 = w0 + HDIM;
    b0.x = w0[0];   b0.y = w1[0];
    b1.x = w0[16];  b1.y = w1[16];
    b2.x = w0[32];  b2.y = w1[32];
    b3.x = w0[48];  b3.y = w1[48];

    acc0 = __builtin_amdgcn_wmma_f32_16x16x4_f32(false, a, false, b0, (short)0, acc0, false, false);
    acc1 = __builtin_amdgcn_wmma_f32_16x16x4_f32(false, a, false, b1, (short)0, acc1, false, false);
    acc2 = __builtin_amdgcn_wmma_f32_16x16x4_f32(false, a, false, b2, (short)0, acc2, false, false);
    acc3 = __builtin_amdgcn_wmma_f32_16x16x4_f32(false, a, false, b3, (short)0, acc3, false, false);
  }

  // D layout: VGPR r holds row (r + hi*8), col = lo (per 16x16 f32 C/D layout)
#pragma unroll
  for (int r = 0; r < 8; ++r) {
    float* yrow = Y + (size_t)(row0 + r + hi * 8) * HDIM + col0 + lo;
    yrow[0]  = acc0[r] * scale;
    yrow[16] = acc1[r] * scale;
    yrow[32] = acc2[r] * scale;
    yrow[48] = acc3[r] * scale;
  }
}

// ---------------------------------------------------------------------------
// logits[b][m][e] = sum_k Mp[b][m][k] * Ep[b][e][k]   (B operand = Ep^T)
// grid = (LENM/16, LENE/256, BS), block = 128 (4 waves, 16x64 strip per wave)
// ---------------------------------------------------------------------------
__global__ void logits_gemm_wmma(const float* __restrict__ Mp,
                                 const float* __restrict__ Ep,
                                 float* __restrict__ Lg) {
  const int lane = threadIdx.x & 31;
  const int wave = threadIdx.x >> 5;
  const int b  = blockIdx.z;
  const int m0 = blockIdx.x * 16;
  const int e0 = blockIdx.y * 256 + wave * 64;
  const int hi = lane >> 4;
  const int lo = lane & 15;

  const float* Mb = Mp + (size_t)b * LENM * HDIM;
  const float* Eb = Ep + (size_t)b * LENE * HDIM;
  float*       Lb = Lg + (size_t)b * LENM * LENE;

  v8f acc0 = {}, acc1 = {}, acc2 = {}, acc3 = {};

  const float* mrow = Mb + (size_t)(m0 + lo) * HDIM;
  const float* e0p  = Eb + (size_t)(e0 + lo)      * HDIM;
  const float* e1p  = Eb + (size_t)(e0 + 16 + lo) * HDIM;
  const float* e2p  = Eb + (size_t)(e0 + 32 + lo) * HDIM;
  const float* e3p  = Eb + (size_t)(e0 + 48 + lo) * HDIM;

  for (int k0 = 0; k0 < HDIM; k0 += 4) {
    const int kk = k0 + 2 * hi;
    v2f a;
    a.x = mrow[kk];
    a.y = mrow[kk + 1];

    v2f b0, b1, b2, b3;
    b0.x = e0p[kk]; b0.y = e0p[kk + 1];
    b1.x = e1p[kk]; b1.y = e1p[kk + 1];
    b2.x = e2p[kk]; b2.y = e2p[kk + 1];
    b3.x = e3p[kk]; b3.y = e3p[kk + 1];

    acc0 = __builtin_amdgcn_wmma_f32_16x16x4_f32(false, a, false, b0, (short)0, acc0, false, false);
    acc1 = __builtin_amdgcn_wmma_f32_16x16x4_f32(false, a, false, b1, (short)0, acc1, false, false);
    acc2 = __builtin_amdgcn_wmma_f32_16x16x4_f32(false, a, false, b2, (short)0, acc2, false, false);
    acc3 = __builtin_amdgcn_wmma_f32_16x16x4_f32(false, a, false, b3, (short)0, acc3, false, false);
  }

#pragma unroll
  for (int r = 0; r < 8; ++r) {
    float* lrow = Lb + (size_t)(m0 + r + hi * 8) * LENE + e0 + lo;
    lrow[0]  = acc0[r];
    lrow[16] = acc1[r];
    lrow[32] = acc2[r];
    lrow[48] = acc3[r];
  }
}

// ---------------------------------------------------------------------------
// Block reductions (blockDim == 512)
// ---------------------------------------------------------------------------
__device__ __forceinline__ float block_reduce_sum(float v, float* s) {
  const int t = threadIdx.x;
  s[t] = v;
  __syncthreads();
#pragma unroll
  for (int off = 256; off >= 1; off >>= 1) {
    if (t < off) s[t] += s[t + off];
    __syncthreads();
  }
  float r = s[0];
  __syncthreads();
  return r;
}

__device__ __forceinline__ float block_reduce_max(float v, float* s) {
  const int t = threadIdx.x;
  s[t] = v;
  __syncthreads();
#pragma unroll
  for (int off = 256; off >= 1; off >>= 1) {
    if (t < off) s[t] = fmaxf(s[t], s[t + off]);
    __syncthreads();
  }
  float r = s[0];
  __syncthreads();
  return r;
}

// ---------------------------------------------------------------------------
// Per (b,m) row: lambda, softmax_q, softmax_c, combine, deterministic scatter.
// grid = BS*LENM blocks of LENE (=512) threads.
// ---------------------------------------------------------------------------
__global__ void softmax_scatter_kernel(const float* __restrict__ Lg,
                                       const float* __restrict__ bq,
                                       const float* __restrict__ bc,
                                       const int*   __restrict__ idx,
                                       const float* __restrict__ D,
                                       const float* __restrict__ Cq,
                                       const float* __restrict__ Cc,
                                       const float* __restrict__ Wl,
                                       const float* __restrict__ bl,
                                       float* __restrict__ out) {
  __shared__ float red[LENE];
  __shared__ int   sidx[LENE];
  __shared__ float sval[LENE];

  const int t   = threadIdx.x;      // e index (and h index for lambda)
  const int row = blockIdx.x;       // b*LENM + m
  const int b   = row / LENM;

  // ---- lambda = sigmoid([D;Cq;Cc] @ W_lambda + b_lambda) ----
  const size_t rh = (size_t)row * HDIM;
  float part = D[rh + t]  * Wl[t]
             + Cq[rh + t] * Wl[HDIM + t]
             + Cc[rh + t] * Wl[2 * HDIM + t];
  float lsum = block_reduce_sum(part, red);
  float lam  = 1.0f / (1.0f + __expf(-(lsum + bl[0])));

  // ---- shared-base-logit softmaxes ----
  float base = Lg[(size_t)row * LENE + t];
  float lq = base + bq[(size_t)b * LENE + t];
  float lc = base + bc[(size_t)b * LENE + t];

  float mq = block_reduce_max(lq, red);
  float eq = __expf(lq - mq);
  float sq = block_reduce_sum(eq, red);

  float mc = block_reduce_max(lc, red);
  float ec = __expf(lc - mc);
  float sc = block_reduce_sum(ec, red);

  float comb = lam * (eq / sq) + (1.0f - lam) * (ec / sc);

  // ---- deterministic scatter-add over vocab (no atomics) ----
  const int v = idx[(size_t)b * LENE + t];
  sidx[t] = v;
  sval[t] = comb;
  __syncthreads();

  bool owner = true;
  for (int e = 0; e < t; ++e) {
    if (sidx[e] == v) { owner = false; break; }
  }
  if (owner) {
    float tot = sval[t];
    for (int e = t + 1; e < LENE; ++e) {
      if (sidx[e] == v) tot += sval[e];
    }
    out[(size_t)row * VOCAB + v] = tot;
  }
}

// ---------------------------------------------------------------------------
extern "C" void kernel_launch(void* const* d_in, const int* in_sizes, int n_in,
                              void* d_out, int out_size, void* d_ws, size_t ws_size,
                              hipStream_t stream) {
  (void)in_sizes; (void)n_in; (void)ws_size;

  const int*   inputs = (const int*)  d_in[0];
  const float* D      = (const float*)d_in[1];
  const float* Cq     = (const float*)d_in[2];
  const float* Cc     = (const float*)d_in[3];
  const float* M      = (const float*)d_in[4];
  const float* E      = (const float*)d_in[5];
  const float* abq    = (const float*)d_in[6];
  const float* abc    = (const float*)d_in[7];
  const float* Wl     = (const float*)d_in[8];
  const float* bl     = (const float*)d_in[9];
  const float* We     = (const float*)d_in[10];
  const float* Wm     = (const float*)d_in[11];
  float* out = (float*)d_out;

  float* ws = (float*)d_ws;
  float* Ep = ws;                                    // BS*LENE*HDIM
  float* Mp = Ep + (size_t)BS * LENE * HDIM;         // BS*LENM*HDIM
  float* Lg = Mp + (size_t)BS * LENM * HDIM;         // BS*LENM*LENE

  // 1) zero the (mostly-sparse) output, non-temporal
  long n4 = (long)out_size / 4;
  zero_out_kernel<<<4096, 256, 0, stream>>>(out, n4);

  // 2) Ep = E @ W_e  (rows = BS*LENE = 2048)
  proj_gemm_wmma<<<dim3((BS * LENE) / 16, HDIM / 256), 128, 0, stream>>>(E, We, Ep, 1.0f);

  // 3) Mp = (M @ W_m) * H^-0.5  (rows = BS*LENM = 1024)
  const float scale = 0.044194173824159216f;  // 512^-0.5
  proj_gemm_wmma<<<dim3((BS * LENM) / 16, HDIM / 256), 128, 0, stream>>>(M, Wm, Mp, scale);

  // 4) logits = Mp @ Ep^T per batch
  logits_gemm_wmma<<<dim3(LENM / 16, LENE / 256, BS), 128, 0, stream>>>(Mp, Ep, Lg);

  // 5) fused lambda + dual softmax + combine + deterministic scatter
  softmax_scatter_kernel<<<BS * LENM, LENE, 0, stream>>>(Lg, abq, abc, inputs,
                                                         D, Cq, Cc, Wl, bl, out);
}